// OneLayerDiagTransitionLinearRNN_60069412601936
// MI455X (gfx1250) — compile-verified
//
#include <hip/hip_runtime.h>

typedef __attribute__((ext_vector_type(2))) float v2f;
typedef __attribute__((ext_vector_type(8))) float v8f;
typedef __attribute__((ext_vector_type(4))) unsigned int v4u;
typedef __attribute__((ext_vector_type(4))) int v4i;
typedef __attribute__((ext_vector_type(8))) int v8i;

// Problem dims (fixed by the reference)
constexpr int B_ = 32, S_ = 4096, I_ = 64, H_ = 512, O_ = 64;
constexpr int L_ = 64;           // scan chunk length (timesteps per workgroup)
constexpr int J_ = S_ / L_;      // 64 chunks along S

// ---- LDS layout (dword offsets). Row pitch ≡ 4 (mod 64) makes every WMMA
// fragment ds_load_b64 bank-conflict free (lanes 0..15 -> banks {4l+k},
// lanes 16..31 (K+2) -> {4l+k+2}).
constexpr int XBP = 516;                  // pitch of xb / h rows   (64 x 516)
constexpr int XP  = 68;                   // pitch of x-chunk rows  (64 x 68)
constexpr int BTP = 68;                   // pitch of bT rows       (512 x 68)
constexpr int XB_OFF = 0;
constexpr int XC_OFF = L_ * XBP;          // 33024
constexpr int BT_OFF = XC_OFF + L_ * XP;  // 37376
constexpr int CT_OFF = XC_OFF;            // cT overlays Xc+bT after GEMM1
constexpr int LDS_DW = BT_OFF + H_ * BTP; // 72192 dwords = 288768 bytes

// ---- Tensor Data Mover: 1-D DMA of `ndw` dwords from contiguous global
// memory into LDS, inserting `padA` encoded dwords of pad every `padI`
// encoded dwords (pad_interval v -> 2^(v+1) dwords, pad_amount v -> v+1).
// Descriptor per CDNA5 ISA (count=1, type=2, dword data).
__device__ __forceinline__ void tdm_load_1d(const void* gp, unsigned lds_byte,
                                            unsigned ndw, int padI, int padA) {
  unsigned long long ga = (unsigned long long)(uintptr_t)gp;
  v4u g0 = { 1u,                                   // count=1, user descriptor
             lds_byte,                             // lds_addr (bytes)
             (unsigned)ga,                         // global_addr[31:0]
             (unsigned)((ga >> 32) & 0x1FFFFFFu) | (2u << 30) }; // ga[56:32] | type=2
  unsigned w0 = (2u << 16)                         // data_size = 4 bytes
              | (1u << 20)                         // pad_enable
              | ((unsigned)padI << 22)             // pad_interval
              | ((unsigned)padA << 25);            // pad_amount
  v8i g1 = { (int)w0,
             (int)((ndw & 0xFFFFu) << 16),         // [47:32]=atomic_addr=0, td0[15:0]
             (int)(((ndw >> 16) & 0xFFFFu) | (1u << 16)), // td0[31:16], td1=1
             (int)((ndw & 0xFFFFu) << 16),         // td1 hi=0, tile_dim0 = ndw
             0,                                    // tile_dim1=0 (1-D), tile_dim2=0
             (int)ndw,                             // tensor_dim0_stride lo32
             0, 0 };
  v4i z = {};
#if defined(__clang_major__) && (__clang_major__ >= 23)
  v8i z8 = {};
  __builtin_amdgcn_tensor_load_to_lds(g0, g1, z, z, z8, 0);
#else
  __builtin_amdgcn_tensor_load_to_lds(g0, g1, z, z, 0);
#endif
}

// Stage x-chunk + bT into LDS via TDM, then GEMM1:
// xb(64x512) = Xc(64x64) @ b(64x512) with V_WMMA_F32_16X16X4_F32.
// Wave w owns M-tile (w&3) and N-strips (w>>2)*8..+7, so its 16 A-fragments
// are loaded once into registers and reused for all 8 strips.
__device__ __forceinline__ void stage_and_gemm1(const float* __restrict__ x,
                                                const float* __restrict__ bT,
                                                float* smem, int bb, int j) {
  const int tid = threadIdx.x;
  float* sXB = smem + XB_OFF;
  const float* sX  = smem + XC_OFF;
  const float* sBT = smem + BT_OFF;

  if (tid < 32) {  // wave 0 drives the TDM engine (EXEC ignored by tensor ops)
    const float* xc = x + ((size_t)bb * S_ + (size_t)j * L_) * I_;
    tdm_load_1d(xc, XC_OFF * 4u, L_ * I_, /*64dw*/5, /*+4dw*/3);  // pitch 68
    tdm_load_1d(bT, BT_OFF * 4u, H_ * I_, 5, 3);                  // pitch 68
    __builtin_amdgcn_s_wait_tensorcnt(0);
  }
  __syncthreads();

  const int wave = tid >> 5, lane = tid & 31;
  const int l15 = lane & 15, hi = lane >> 4;
  const int m0 = (wave & 3) * 16;
  const int g  = wave >> 2;

  v2f afr[16];                      // A-fragments for all 16 k-steps, reused 8x
  #pragma unroll
  for (int kk = 0; kk < 16; ++kk)
    afr[kk] = *(const v2f*)&sX[(m0 + l15) * XP + 4 * kk + 2 * hi];

  #pragma unroll
  for (int q = 0; q < 8; ++q) {
    const int n0 = (g * 8 + q) * 16;
    v8f acc = {};
    #pragma unroll
    for (int kk = 0; kk < 16; ++kk) {
      v2f bv = *(const v2f*)&sBT[(n0 + l15) * BTP + 4 * kk + 2 * hi];
      acc = __builtin_amdgcn_wmma_f32_16x16x4_f32(false, afr[kk], false, bv,
                                                  (short)0, acc, false, false);
    }
    const int base = (m0 + 8 * hi) * XBP + n0 + l15;
    #pragma unroll
    for (int r = 0; r < 8; ++r) sXB[base + r * XBP] = acc[r];
  }
  __syncthreads();
}

// Pass 1: per-chunk local scan end state (h after the chunk, entering with 0).
__global__ void __launch_bounds__(512)
k_chunk_states(const float* __restrict__ x, const float* __restrict__ bT,
               const float* __restrict__ A, float* __restrict__ states) {
  extern __shared__ float smem[];
  const int bb = blockIdx.x / J_;
  const int j  = blockIdx.x % J_;
  stage_and_gemm1(x, bT, smem, bb, j);
  const int h = threadIdx.x;
  const float a = A[h];
  const float* sXB = smem + XB_OFF;
  float s = 0.f;
  #pragma unroll
  for (int t = 0; t < L_; ++t) s = fmaf(s, a, sXB[t * XBP + h]);
  states[((size_t)bb * J_ + j) * H_ + h] = s;
}

// Pass 2: tiny cross-chunk exclusive scan (in place):
// states[j] becomes the carry entering chunk j; carry' = A^64*carry + end_j.
__global__ void __launch_bounds__(512)
k_carry_scan(const float* __restrict__ A, float* __restrict__ states) {
  const int h  = threadIdx.x;
  const int bb = blockIdx.x;
  float a = A[h];
  float aL = a;
  #pragma unroll
  for (int q = 0; q < 6; ++q) aL *= aL;      // a^64 by repeated squaring
  float carry = 0.f;
  size_t base = ((size_t)bb * J_) * H_ + h;
  for (int jj = 0; jj < J_; ++jj) {
    float s = states[base + (size_t)jj * H_];
    states[base + (size_t)jj * H_] = carry;
    carry = fmaf(carry, aL, s);
  }
}

// Pass 3: recompute xb, scan with carry-in (in LDS), GEMM2 with c, store y.
// cT is DMA'd (overlapping the scan) with 2-dword pads every 256 dwords:
// element (o,h) lives at o*516 + h + 2*(h>=256).
__global__ void __launch_bounds__(512)
k_output(const float* __restrict__ x, const float* __restrict__ bT,
         const float* __restrict__ cT, const float* __restrict__ A,
         const float* __restrict__ carry, float* __restrict__ out) {
  extern __shared__ float smem[];
  const int bb = blockIdx.x / J_;
  const int j  = blockIdx.x % J_;
  stage_and_gemm1(x, bT, smem, bb, j);

  float* sXB = smem + XB_OFF;
  const float* sCT = smem + CT_OFF;
  const int tid = threadIdx.x;

  if (tid < 32)  // overlay region is free after the GEMM1 barrier
    tdm_load_1d(cT, CT_OFF * 4u, O_ * H_, /*256dw*/7, /*+2dw*/1);

  // in-place diagonal scan along t for channel h = tid (disjoint from sCT),
  // overlapped with the cT tensor DMA
  {
    const int h = tid;
    const float a = A[h];
    float st = carry[((size_t)bb * J_ + j) * H_ + h];
    #pragma unroll
    for (int t = 0; t < L_; ++t) {
      st = fmaf(st, a, sXB[t * XBP + h]);
      sXB[t * XBP + h] = st;
    }
  }
  if (tid < 32) __builtin_amdgcn_s_wait_tensorcnt(0);
  __syncthreads();

  // GEMM2: Y(64x64) = H(64x512) @ c(512x64); 16 waves, one 16x16 tile each
  const int wave = tid >> 5, lane = tid & 31;
  const int l15 = lane & 15, hi = lane >> 4;
  const int m0 = (wave & 3) * 16, n0 = (wave >> 2) * 16;
  v8f acc = {};
  #pragma unroll
  for (int k = 0; k < 256; k += 4) {
    v2f av = *(const v2f*)&sXB[(m0 + l15) * XBP + k + 2 * hi];
    v2f bv = *(const v2f*)&sCT[(n0 + l15) * 516 + k + 2 * hi];
    acc = __builtin_amdgcn_wmma_f32_16x16x4_f32(false, av, false, bv,
                                                (short)0, acc, false, false);
  }
  #pragma unroll
  for (int k = 256; k < 512; k += 4) {
    v2f av = *(const v2f*)&sXB[(m0 + l15) * XBP + k + 2 * hi];
    v2f bv = *(const v2f*)&sCT[(n0 + l15) * 516 + k + 2 * hi + 2];
    acc = __builtin_amdgcn_wmma_f32_16x16x4_f32(false, av, false, bv,
                                                (short)0, acc, false, false);
  }
  // out flat index = bb*S*O + (j*64 + t)*O + o
  float* op = out + ((size_t)bb * S_ + (size_t)j * L_) * O_;
  const int rbase = m0 + 8 * hi;
  #pragma unroll
  for (int r = 0; r < 8; ++r) op[(rbase + r) * O_ + n0 + l15] = acc[r];
}

// Pre-transpose b (64x512 -> bT 512x64) and c (512x64 -> cT 64x512) so WMMA
// fragment loads have contiguous K.
__global__ void __launch_bounds__(512)
k_pack(const float* __restrict__ b, const float* __restrict__ c,
       float* __restrict__ bT, float* __restrict__ cT) {
  int i = blockIdx.x * blockDim.x + threadIdx.x;
  if (i < I_ * H_) {
    int h = i >> 6, ii = i & 63;     // bT[h][ii] = b[ii][h]
    bT[i] = b[ii * H_ + h];
    int o = i >> 9, hh = i & 511;    // cT[o][hh] = c[hh][o]
    cT[i] = c[hh * O_ + o];
  }
}

extern "C" void kernel_launch(void* const* d_in, const int* in_sizes, int n_in,
                              void* d_out, int out_size, void* d_ws, size_t ws_size,
                              hipStream_t stream) {
  const float* x = (const float*)d_in[0];   // (32, 4096, 64)
  const float* b = (const float*)d_in[1];   // (64, 512)
  const float* A = (const float*)d_in[2];   // (512,)
  const float* c = (const float*)d_in[3];   // (512, 64)
  float* out = (float*)d_out;               // 32*4096*64 fp32

  float* ws     = (float*)d_ws;             // uses ~4.5 MB of scratch
  float* bT     = ws;                        // 32768 floats
  float* cT     = ws + 32768;                // 32768 floats
  float* states = ws + 65536;                // B*J*H = 1,048,576 floats

  constexpr size_t shmem = (size_t)LDS_DW * sizeof(float);  // 288768 B

  k_pack<<<64, 512, 0, stream>>>(b, c, bT, cT);
  k_chunk_states<<<B_ * J_, 512, shmem, stream>>>(x, bT, A, states);
  k_carry_scan<<<B_, 512, 0, stream>>>(A, states);
  k_output<<<B_ * J_, 512, shmem, stream>>>(x, bT, cT, A, states, out);
}